// BornMayerHugginsForce_50757923504450
// MI455X (gfx1250) — compile-verified
//
#include <hip/hip_runtime.h>
#include <stdint.h>

#define NATOMS 4096
#define TPB    256
#define TILE   512
#define NTILES (NATOMS / TILE)   // 8
#define PF_OPS 8                 // async-load instructions per prefetch (per wave)

// ---- CDNA5 async global->LDS copy (guarded builtin, inline-asm fallback) ----
#if defined(__has_builtin)
# if __has_builtin(__builtin_amdgcn_global_load_async_to_lds_b64)
#  define HAVE_ASYNC_BUILTIN 1
# endif
# if __has_builtin(__builtin_amdgcn_s_wait_asynccnt)
#  define HAVE_WAIT_BUILTIN 1
# endif
#endif

// gfx1250 async builtins are typed (v2i AS1*, v2i AS3*, imm, imm)
typedef int v2i __attribute__((ext_vector_type(2)));
typedef __attribute__((address_space(1))) v2i* g_v2i_ptr;   // global, 64-bit
typedef __attribute__((address_space(3))) v2i* l_v2i_ptr;   // LDS, 32-bit

__device__ __forceinline__ void async_copy8(const float* g, float* l) {
#if defined(HAVE_ASYNC_BUILTIN)
  __builtin_amdgcn_global_load_async_to_lds_b64(
      (g_v2i_ptr)(uintptr_t)g,
      (l_v2i_ptr)(uint32_t)(uintptr_t)l,
      0, 0);
#else
  uint32_t lo = (uint32_t)(uintptr_t)l;   // low 32 bits of generic LDS addr
  asm volatile("global_load_async_to_lds_b64 %0, %1, off"
               :: "v"(lo), "v"(g) : "memory");
#endif
}

#if defined(HAVE_WAIT_BUILTIN)
# define WAIT_ASYNC(n) __builtin_amdgcn_s_wait_asynccnt(n)
#else
# define WAIT_ASYNC(n) asm volatile("s_wait_asynccnt %0" :: "n"(n) : "memory")
#endif

// ---------------------------------------------------------------------------
// One block per row i. Streams A/C/D/rho/sigma row tiles (+coord slice) into
// LDS with double-buffered async loads; computes pair energy + analytic force.
// ---------------------------------------------------------------------------
__global__ __launch_bounds__(TPB)
void bmh_kernel(const float* __restrict__ coords,
                const float* __restrict__ Ap, const float* __restrict__ Cp,
                const float* __restrict__ Dp, const float* __restrict__ Rp,
                const float* __restrict__ Sp,
                const float* __restrict__ box,
                const int* __restrict__ cut_p, const int* __restrict__ md_p,
                float* __restrict__ out, float* __restrict__ e_part)
{
  __shared__ float s_par[2][5][TILE];    // 20 KB
  __shared__ float s_crd[2][3 * TILE];   // 12 KB

  const int i = blockIdx.x;
  const int t = threadIdx.x;

  const size_t row = (size_t)i * NATOMS;
  const float* rA = Ap + row;
  const float* rC = Cp + row;
  const float* rD = Dp + row;
  const float* rR = Rp + row;
  const float* rS = Sp + row;

  const float Lx = box[0], Ly = box[1], Lz = box[2];
  const float iLx = 1.0f / Lx, iLy = 1.0f / Ly, iLz = 1.0f / Lz;
  const float cut = (float)cut_p[0];
  const float md  = (float)md_p[0];
  const float md2 = md * md;

  const float xi = coords[3 * i + 0];
  const float yi = coords[3 * i + 1];
  const float zi = coords[3 * i + 2];

  auto issue = [&](int b, int k) {
    const int g0 = k * TILE + 2 * t;               // float index into the row
    async_copy8(rA + g0, &s_par[b][0][2 * t]);
    async_copy8(rC + g0, &s_par[b][1][2 * t]);
    async_copy8(rD + g0, &s_par[b][2][2 * t]);
    async_copy8(rR + g0, &s_par[b][3][2 * t]);
    async_copy8(rS + g0, &s_par[b][4][2 * t]);
    const float* gc = coords + 3 * (k * TILE);     // 1536 contiguous floats
    async_copy8(gc + 2 * t,            &s_crd[b][2 * t]);
    async_copy8(gc + TILE + 2 * t,     &s_crd[b][TILE + 2 * t]);
    async_copy8(gc + 2 * TILE + 2 * t, &s_crd[b][2 * TILE + 2 * t]);
  };

  issue(0, 0);

  float fx = 0.f, fy = 0.f, fz = 0.f, esum = 0.f;

  for (int k = 0; k < NTILES; ++k) {
    const int b = k & 1;
    if (k + 1 < NTILES) { issue(b ^ 1, k + 1); WAIT_ASYNC(PF_OPS); }
    else                { WAIT_ASYNC(0); }
    __syncthreads();

#pragma unroll
    for (int m = 0; m < TILE / TPB; ++m) {
      const int jj = t + m * TPB;
      const int j  = k * TILE + jj;

      const float xj = s_crd[b][3 * jj + 0];
      const float yj = s_crd[b][3 * jj + 1];
      const float zj = s_crd[b][3 * jj + 2];

      float dx = xi - xj; dx -= rintf(dx * iLx) * Lx;   // minimum image (rne)
      float dy = yi - yj; dy -= rintf(dy * iLy) * Ly;
      float dz = zi - zj; dz -= rintf(dz * iLz) * Lz;

      const float r2  = fmaf(dx, dx, fmaf(dy, dy, dz * dz));
      const bool  gok = (r2 >= md2);                     // clamp-grad gate
      const float r2c = fmaxf(r2, md2);
      const float ivr = __builtin_amdgcn_rsqf(r2c);
      const float r   = r2c * ivr;

      if ((j != i) && (r < cut)) {
        const float a  = s_par[b][0][jj];
        const float c  = s_par[b][1][jj];
        const float d  = s_par[b][2][jj];
        const float rh = s_par[b][3][jj];
        const float sg = s_par[b][4][jj];

        const float ivr2 = ivr * ivr;
        const float ivr6 = ivr2 * ivr2 * ivr2;
        const float ivr8 = ivr6 * ivr2;
        const float irho = __builtin_amdgcn_rcpf(rh);
        const float ex   = a * __expf((sg - r) * irho);

        esum += fmaf(d, ivr8, fmaf(-c, ivr6, ex));       // A e^{(s-r)/p} - C/r^6 + D/r^8

        if (gok) {
          // de/dr = -ex/rho + (6C/r^6 - 8D/r^8)/r ;  F_i += -de/dr * (dx,dy,dz)/r
          const float de = fmaf(-ex, irho,
                                (6.0f * c * ivr6 - 8.0f * d * ivr8) * ivr);
          const float s  = -de * ivr;
          fx = fmaf(s, dx, fx);
          fy = fmaf(s, dy, fy);
          fz = fmaf(s, dz, fz);
        }
      }
    }
    __syncthreads();
  }

  // fixed-order block reduction (deterministic) reusing s_par as scratch
  float* red = &s_par[0][0][0];
  red[t          ] = fx;
  red[t +   TPB  ] = fy;
  red[t + 2 * TPB] = fz;
  red[t + 3 * TPB] = esum;
  __syncthreads();
  for (int s = TPB >> 1; s > 0; s >>= 1) {
    if (t < s) {
      red[t          ] += red[t           + s];
      red[t +   TPB  ] += red[t +   TPB   + s];
      red[t + 2 * TPB] += red[t + 2 * TPB + s];
      red[t + 3 * TPB] += red[t + 3 * TPB + s];
    }
    __syncthreads();
  }
  if (t == 0) {
    out[1 + 3 * i + 0] = red[0];
    out[1 + 3 * i + 1] = red[TPB];
    out[1 + 3 * i + 2] = red[2 * TPB];
    e_part[i] = red[3 * TPB];
  }
}

// Deterministic final energy reduction: fixed per-thread stride order + tree.
__global__ __launch_bounds__(256)
void bmh_reduce(const float* __restrict__ e_part, float* __restrict__ out) {
  __shared__ float s[256];
  const int t = threadIdx.x;
  float acc = 0.f;
  for (int k = t; k < NATOMS; k += 256) acc += e_part[k];
  s[t] = acc;
  __syncthreads();
  for (int w = 128; w > 0; w >>= 1) {
    if (t < w) s[t] += s[t + w];
    __syncthreads();
  }
  if (t == 0) out[0] = 0.5f * s[0];
}

extern "C" void kernel_launch(void* const* d_in, const int* in_sizes, int n_in,
                              void* d_out, int out_size, void* d_ws, size_t ws_size,
                              hipStream_t stream) {
  (void)in_sizes; (void)n_in; (void)out_size; (void)ws_size;
  const float* coords = (const float*)d_in[0];
  const float* A      = (const float*)d_in[1];
  const float* C      = (const float*)d_in[2];
  const float* D      = (const float*)d_in[3];
  const float* rho    = (const float*)d_in[4];
  const float* sigma  = (const float*)d_in[5];
  const float* box    = (const float*)d_in[6];
  const int*   cutoff = (const int*)d_in[7];
  const int*   mind   = (const int*)d_in[8];
  float* out   = (float*)d_out;
  float* epart = (float*)d_ws;   // 4096 floats = 16 KB scratch

  bmh_kernel<<<NATOMS, TPB, 0, stream>>>(coords, A, C, D, rho, sigma,
                                         box, cutoff, mind, out, epart);
  bmh_reduce<<<1, 256, 0, stream>>>(epart, out);
}